// ATLModule_8770323218766
// MI455X (gfx1250) — compile-verified
//
#include <hip/hip_runtime.h>
#include <hip/hip_bf16.h>

typedef __attribute__((ext_vector_type(2))) float v2f;
typedef __attribute__((ext_vector_type(8))) float v8f;

#define Q   64
#define S   25
#define C   640
#define HW  100
#define M   (Q*HW)     // 6400 rows (query pixels)
#define N   (S*HW)     // 2500 cols (support pixels)
#define NP  2528       // N padded to multiple of 32 (79 blocks of 2 tiles)
#define CH  40         // C/16 hidden units

// ---------------------------------------------------------------------------
// K1: per-query-pixel L2 normalize + AEA threshold MLP
// qdata: [Q][C][HW] -> qn: [M][C] row-major, cv: [M]
// ---------------------------------------------------------------------------
__global__ void __launch_bounds__(256) qnorm_mlp_kernel(
    const float* __restrict__ qdata,
    const float* __restrict__ w1, const float* __restrict__ b1,
    const float* __restrict__ w2, const float* __restrict__ b2,
    float* __restrict__ qn, float* __restrict__ cv)
{
    __shared__ float sh[C];
    __shared__ float red[256];
    __shared__ float hid[CH];
    __shared__ float inv_s;

    const int m  = blockIdx.x;          // query pixel row
    const int qi = m / HW, px = m % HW;
    const int t  = threadIdx.x;
    const float* base = qdata + (size_t)qi * C * HW + px;

    float ss = 0.0f;
    for (int ch = t; ch < C; ch += 256) {
        float v = base[(size_t)ch * HW];
        sh[ch] = v;
        ss += v * v;
    }
    red[t] = ss; __syncthreads();
    for (int w = 128; w > 0; w >>= 1) { if (t < w) red[t] += red[t + w]; __syncthreads(); }
    if (t == 0) inv_s = 1.0f / fmaxf(sqrtf(red[0]), 1e-12f);
    __syncthreads();

    const float inv = inv_s;
    for (int ch = t; ch < C; ch += 256) {
        float v = sh[ch] * inv;
        sh[ch] = v;
        qn[(size_t)m * C + ch] = v;
    }
    __syncthreads();

    // hid = leaky_relu(qn . w1^T + b1, 0.2)
    if (t < CH) {
        const float* wr = w1 + (size_t)t * C;
        float acc = b1[t];
        for (int ch = 0; ch < C; ++ch) acc = fmaf(sh[ch], wr[ch], acc);
        hid[t] = acc > 0.0f ? acc : 0.2f * acc;
    }
    __syncthreads();

    if (t == 0) {
        float acc = b2[0];
        for (int j = 0; j < CH; ++j) acc = fmaf(hid[j], w2[j], acc);
        float psi = 1.0f / (1.0f + expf(-acc));
        cv[m] = psi * 0.3f + 0.5f;
    }
}

// ---------------------------------------------------------------------------
// K2: per-support-pixel L2 normalize -> snp: [C][NP] (pad cols zeroed)
// sdata: [S][C][HW]
// ---------------------------------------------------------------------------
__global__ void __launch_bounds__(256) snorm_kernel(
    const float* __restrict__ sdata, float* __restrict__ snp)
{
    __shared__ float red[256];
    __shared__ float inv_s;

    const int n = blockIdx.x;           // 0..NP-1
    const int t = threadIdx.x;

    if (n >= N) {                       // zero padding columns
        for (int ch = t; ch < C; ch += 256) snp[(size_t)ch * NP + n] = 0.0f;
        return;
    }
    const int si = n / HW, px = n % HW;
    const float* base = sdata + (size_t)si * C * HW + px;

    float ss = 0.0f;
    for (int ch = t; ch < C; ch += 256) {
        float v = base[(size_t)ch * HW];
        ss += v * v;
    }
    red[t] = ss; __syncthreads();
    for (int w = 128; w > 0; w >>= 1) { if (t < w) red[t] += red[t + w]; __syncthreads(); }
    if (t == 0) inv_s = 1.0f / fmaxf(sqrtf(red[0]), 1e-12f);
    __syncthreads();

    const float inv = inv_s;
    for (int ch = t; ch < C; ch += 256)
        snp[(size_t)ch * NP + n] = base[(size_t)ch * HW] * inv;
}

// ---------------------------------------------------------------------------
// K3: f_x = qn @ snp with V_WMMA_F32_16X16X4_F32, register-blocked 4Mx2N.
// One wave computes a 64x32 output block: 8 accumulators, and per K-step of 4
// issues 8 wmma from 4 A-fragments + 2 B-fragments (~1 wmma per vmem load).
// Fragment layouts (ISA 7.12.2):
//   A (16x4 f32): lane l holds M=l&15, VGPR v holds K = 2*(l>>4)+v -> float2
//   B (4x16 f32): lane l holds N=l&15, VGPR v holds K = 2*(l>>4)+v
//   D (16x16 f32): lane l holds N=l&15, VGPR v holds M = v + 8*(l>>4)
// ---------------------------------------------------------------------------
__global__ void __launch_bounds__(32) gemm_wmma_f32(
    const float* __restrict__ qn,
    const float* __restrict__ snp,
    float* __restrict__ fx)
{
    const int lane = threadIdx.x;
    const int m0 = blockIdx.x * 64;     // 4 M-tiles
    const int n0 = blockIdx.y * 32;     // 2 N-tiles
    const int r  = lane & 15;
    const int hi = lane >> 4;

    v8f acc[4][2];
    #pragma unroll
    for (int i = 0; i < 4; ++i)
        #pragma unroll
        for (int j = 0; j < 2; ++j)
            acc[i][j] = (v8f){};

    const float* arow0 = qn  + (size_t)(m0 + r) * C + 2 * hi;
    const float* bcol0 = snp + (size_t)(2 * hi) * NP + n0 + r;

    #pragma unroll 2
    for (int k0 = 0; k0 < C; k0 += 4) {
        v2f a[4];
        #pragma unroll
        for (int i = 0; i < 4; ++i)
            a[i] = *(const v2f*)(arow0 + (size_t)i * 16 * C + k0);   // K=k0+2hi,+1

        v2f b[2];
        #pragma unroll
        for (int j = 0; j < 2; ++j) {
            const float* bc = bcol0 + j * 16;
            b[j].x = bc[(size_t)k0 * NP];
            b[j].y = bc[(size_t)(k0 + 1) * NP];
        }

        #pragma unroll
        for (int i = 0; i < 4; ++i)
            #pragma unroll
            for (int j = 0; j < 2; ++j)
                acc[i][j] = __builtin_amdgcn_wmma_f32_16x16x4_f32(
                    false, a[i], false, b[j],
                    (short)0, acc[i][j], false, false);
    }

    #pragma unroll
    for (int i = 0; i < 4; ++i) {
        float* drow = fx + (size_t)(m0 + i * 16 + 8 * hi) * NP + n0 + r;
        #pragma unroll
        for (int j = 0; j < 2; ++j)
            #pragma unroll
            for (int v = 0; v < 8; ++v)
                drow[(size_t)v * NP + j * 16] = acc[i][j][v];
    }
}

// ---------------------------------------------------------------------------
// K4: per-row adaptive attention + fused reduce -> final_local_score
// ---------------------------------------------------------------------------
__global__ void __launch_bounds__(256) epilogue_kernel(
    const float* __restrict__ fx, const float* __restrict__ cv,
    float* __restrict__ flocal)
{
    __shared__ float p[N];
    __shared__ float red[256];
    __shared__ float inv_s;

    const int m = blockIdx.x;
    const int t = threadIdx.x;
    const float c = cv[m];
    const float* row = fx + (size_t)m * NP;

    float ssum = 0.0f;
    for (int n = t; n < N; n += 256) {
        float v  = row[n];
        float cf = 1.0f / (1.0f + expf(-50.0f * (v - c)));
        p[n] = cf * v;
        ssum += fabsf(cf);
    }
    red[t] = ssum; __syncthreads();
    for (int w = 128; w > 0; w >>= 1) { if (t < w) red[t] += red[t + w]; __syncthreads(); }
    if (t == 0) inv_s = 1.0f / fmaxf(red[0], 1e-12f);
    __syncthreads();

    const float inv = inv_s;
    const int qi = m / HW, hq = m % HW;
    for (int si = t; si < S; si += 256) {
        float acc = 0.0f;
        const float* pp = p + si * HW;
        for (int j = 0; j < HW; ++j) acc += pp[j];
        flocal[(size_t)qi * S * HW + (size_t)si * HW + hq] = acc * inv;
    }
}

// ---------------------------------------------------------------------------
// K5: final_score[q,s] = mean_hw(final_local) * 30
// ---------------------------------------------------------------------------
__global__ void __launch_bounds__(256) fscore_kernel(
    const float* __restrict__ flocal, float* __restrict__ fscore)
{
    const int idx = blockIdx.x * blockDim.x + threadIdx.x;
    if (idx >= Q * S) return;
    const float* p = flocal + (size_t)idx * HW;
    float acc = 0.0f;
    for (int j = 0; j < HW; ++j) acc += p[j];
    fscore[idx] = acc * (30.0f / 100.0f);
}

// ---------------------------------------------------------------------------
extern "C" void kernel_launch(void* const* d_in, const int* in_sizes, int n_in,
                              void* d_out, int out_size, void* d_ws, size_t ws_size,
                              hipStream_t stream)
{
    const float* qdata = (const float*)d_in[0];
    const float* sdata = (const float*)d_in[1];
    const float* w1    = (const float*)d_in[2];
    const float* b1    = (const float*)d_in[3];
    const float* w2    = (const float*)d_in[4];
    const float* b2    = (const float*)d_in[5];

    float* out    = (float*)d_out;
    float* fscore = out;             // [Q*S]
    float* flocal = out + Q * S;     // [Q*S*HW]

    // workspace layout (floats): qn[M*C] | cv[M] | snp[C*NP] | fx[M*NP]
    float* ws  = (float*)d_ws;
    float* qn  = ws;
    float* cv  = qn  + (size_t)M * C;
    float* snp = cv  + (size_t)M;
    float* fx  = snp + (size_t)C * NP;

    qnorm_mlp_kernel<<<M, 256, 0, stream>>>(qdata, w1, b1, w2, b2, qn, cv);
    snorm_kernel<<<NP, 256, 0, stream>>>(sdata, snp);
    gemm_wmma_f32<<<dim3(M / 64, NP / 32), 32, 0, stream>>>(qn, snp, fx);
    epilogue_kernel<<<M, 256, 0, stream>>>(fx, cv, flocal);
    fscore_kernel<<<(Q * S + 255) / 256, 256, 0, stream>>>(flocal, fscore);
}